// GCNEncoder_56856777064619
// MI455X (gfx1250) — compile-verified
//
#include <hip/hip_runtime.h>

#define N_NODES 100000
#define N_EDGES 1600000
#define DIN 128
#define DH  128
#define DOUT 64

typedef __attribute__((ext_vector_type(2))) float v2f;
typedef __attribute__((ext_vector_type(8))) float v8f;

// Guaranteed-native fp32 atomic add (global_atomic_add_f32, no CAS loop).
// Safe: all targets are coarse-grained hipMalloc'd device memory.
__device__ __forceinline__ void atomAddF32(float* p, float v) {
    unsafeAtomicAdd(p, v);
}

// ---------------- degree / normalization ----------------

__global__ void k_init_deg(float* __restrict__ deg) {
    int i = blockIdx.x * blockDim.x + threadIdx.x;
    if (i < N_NODES) deg[i] = 1.0f;   // self-loop included
}

__global__ void k_deg_accum(const int* __restrict__ dst, float* __restrict__ deg) {
    int e = blockIdx.x * blockDim.x + threadIdx.x;
    if (e < N_EDGES) atomAddF32(&deg[dst[e]], 1.0f);
}

__global__ void k_dinv(float* __restrict__ deg_dinv) {
    int i = blockIdx.x * blockDim.x + threadIdx.x;
    if (i < N_NODES) deg_dinv[i] = rsqrtf(deg_dinv[i]);
}

__global__ void k_norm_e(const int* __restrict__ src, const int* __restrict__ dst,
                         const float* __restrict__ dinv, float* __restrict__ norme) {
    int e = blockIdx.x * blockDim.x + threadIdx.x;
    if (e < N_EDGES) norme[e] = dinv[src[e]] * dinv[dst[e]];
}

// ---------------- WMMA fp32 GEMM:  out[N x NCOLS] = A[N x K] @ W[K x NCOLS] ----------------
// One wave computes one 16x16 output tile. Grid is sized exactly: no divergent
// early-exit, EXEC stays all-ones around the WMMA (ISA requirement).
// A-operand layout (16x4 f32): lanes 0-15 -> M=lane, K={k,k+1}; lanes 16-31 -> K={k+2,k+3}.
// B-operand layout (4x16 f32): v0/v1 = rows {k,k+1} (lanes 0-15) / {k+2,k+3} (lanes 16-31).
// C/D layout: VGPR r: lanes 0-15 -> row r, lanes 16-31 -> row r+8; col = lane&15.

template<int K, int NCOLS, bool RELU>
__global__ void k_gemm_wmma(const float* __restrict__ A,
                            const float* __restrict__ W,
                            float* __restrict__ out) {
    const int NT = NCOLS / 16;
    int tid  = blockIdx.x * blockDim.x + threadIdx.x;
    int wave = tid >> 5;
    int lane = threadIdx.x & 31;
    int mTile = wave / NT;
    int nTile = wave - mTile * NT;

    int hi      = lane >> 4;      // 0: lanes 0-15, 1: lanes 16-31
    int laneLow = lane & 15;
    int mBase = mTile * 16;
    int nBase = nTile * 16;

    const float* aRow = A + (size_t)(mBase + laneLow) * K;
    const float* wCol = W + nBase + laneLow;

    v8f c = {0.f, 0.f, 0.f, 0.f, 0.f, 0.f, 0.f, 0.f};

    #pragma unroll 4
    for (int k = 0; k < K; k += 4) {
        int kk = k + 2 * hi;
        float a0 = aRow[kk];
        float a1 = aRow[kk + 1];
        if (RELU) { a0 = fmaxf(a0, 0.0f); a1 = fmaxf(a1, 0.0f); }
        v2f av; av.x = a0; av.y = a1;

        v2f bv;
        bv.x = wCol[(size_t)kk * NCOLS];
        bv.y = wCol[(size_t)(kk + 1) * NCOLS];

        // 8 args: (neg_a, A, neg_b, B, c_mod, C, reuse_a, reuse_b)
        c = __builtin_amdgcn_wmma_f32_16x16x4_f32(false, av, false, bv,
                                                  (short)0, c, false, false);
    }

    float* oCol = out + (size_t)(mBase + 8 * hi) * NCOLS + nBase + laneLow;
    #pragma unroll
    for (int r = 0; r < 8; ++r) {
        oCol[(size_t)r * NCOLS] = c[r];
    }
}

// ---------------- epilogue init: agg = h * dinv^2 + bias ----------------

template<int D>
__global__ void k_self_bias(const float* __restrict__ h,
                            const float* __restrict__ dinv,
                            const float* __restrict__ bias,
                            float* __restrict__ agg) {
    int t = blockIdx.x * blockDim.x + threadIdx.x;
    if (t < N_NODES * D) {
        int i = t / D;          // D is a power of two -> shift
        int f = t - i * D;
        float sn = dinv[i] * dinv[i];
        agg[t] = h[t] * sn + bias[f];
    }
}

// ---------------- edge scatter: agg[dst] += h[src] * norm_e ----------------
// One thread per (edge, 4-feature chunk): float4 gather, 4 native fp32 atomics.
// h (51 MB layer1 / 26 MB layer2) and agg fit in the 192 MB L2, so the random
// traffic is L2-resident.

template<int D>
__global__ void k_scatter(const int* __restrict__ src, const int* __restrict__ dst,
                          const float* __restrict__ norme,
                          const float* __restrict__ h, float* __restrict__ agg) {
    const int CH = D / 4;
    int t = blockIdx.x * blockDim.x + threadIdx.x;
    if (t >= N_EDGES * CH) return;
    int e = t / CH;             // power of two -> shift
    int c = t - e * CH;

    int s = src[e], d = dst[e];
    float w = norme[e];

    const float4 v = *(const float4*)(h + (size_t)s * D + c * 4);
    float* p = agg + (size_t)d * D + c * 4;
    atomAddF32(p + 0, v.x * w);
    atomAddF32(p + 1, v.y * w);
    atomAddF32(p + 2, v.z * w);
    atomAddF32(p + 3, v.w * w);
}

// ---------------- launch ----------------

extern "C" void kernel_launch(void* const* d_in, const int* in_sizes, int n_in,
                              void* d_out, int out_size, void* d_ws, size_t ws_size,
                              hipStream_t stream) {
    const float* x   = (const float*)d_in[0];
    const int*   ei  = (const int*)d_in[1];       // [2, E] flat
    const float* W1  = (const float*)d_in[2];
    const float* b1  = (const float*)d_in[3];
    const float* W2  = (const float*)d_in[4];
    const float* b2  = (const float*)d_in[5];
    float*       out = (float*)d_out;

    const int* src = ei;
    const int* dst = ei + N_EDGES;

    // workspace layout (floats)
    float* ws    = (float*)d_ws;
    float* dinv  = ws;                                  // N
    float* norme = dinv  + N_NODES;                     // E
    float* h1    = norme + N_EDGES;                     // N*DH
    float* agg1  = h1    + (size_t)N_NODES * DH;        // N*DH
    float* h2    = agg1  + (size_t)N_NODES * DH;        // N*DOUT

    const int B = 256;

    // 1) degree + normalization
    k_init_deg <<<(N_NODES + B - 1) / B, B, 0, stream>>>(dinv);
    k_deg_accum<<<(N_EDGES + B - 1) / B, B, 0, stream>>>(dst, dinv);
    k_dinv     <<<(N_NODES + B - 1) / B, B, 0, stream>>>(dinv);
    k_norm_e   <<<(N_EDGES + B - 1) / B, B, 0, stream>>>(src, dst, dinv, norme);

    // 2) layer 1: h1 = x @ W1   (WMMA; 6250 * 8 = 50000 waves, exact)
    {
        int waves = (N_NODES / 16) * (DH / 16);
        k_gemm_wmma<DIN, DH, false><<<waves * 32 / B, B, 0, stream>>>(x, W1, h1);
    }
    // 3) agg1 = h1 * dinv^2 + b1
    k_self_bias<DH><<<(N_NODES * DH + B - 1) / B, B, 0, stream>>>(h1, dinv, b1, agg1);
    // 4) agg1[dst] += h1[src] * norm_e
    {
        int total = N_EDGES * (DH / 4);
        k_scatter<DH><<<(total + B - 1) / B, B, 0, stream>>>(src, dst, norme, h1, agg1);
    }

    // 5) layer 2: h2 = relu(agg1) @ W2   (ReLU fused into A loads; 25000 waves, exact)
    {
        int waves = (N_NODES / 16) * (DOUT / 16);
        k_gemm_wmma<DH, DOUT, true><<<waves * 32 / B, B, 0, stream>>>(agg1, W2, h2);
    }
    // 6) out = h2 * dinv^2 + b2, then out[dst] += h2[src] * norm_e
    k_self_bias<DOUT><<<(N_NODES * DOUT + B - 1) / B, B, 0, stream>>>(h2, dinv, b2, out);
    {
        int total = N_EDGES * (DOUT / 4);
        k_scatter<DOUT><<<(total + B - 1) / B, B, 0, stream>>>(src, dst, norme, h2, out);
    }
}